// CellNet_55456617725966
// MI455X (gfx1250) — compile-verified
//
#include <hip/hip_runtime.h>
#include <hip/hip_bf16.h>
#include <math.h>

#define TOPK 700
#define NG 4
#define EDIM 8
#define HW 4096        // 64*64
#define BDIM 2
#define OUT_HW 65536   // 256*256

typedef float v2f __attribute__((ext_vector_type(2)));
typedef float v4f __attribute__((ext_vector_type(4)));
typedef float v8f __attribute__((ext_vector_type(8)));

// sigmoid via v_exp_f32 + v_rcp_f32 (no IEEE divide sequence)
__device__ __forceinline__ float sigmoid_fast(float x) {
    return __builtin_amdgcn_rcpf(1.0f + __expf(-x));
}

// ---------------------------------------------------------------------------
// Kernel A: sigmoid(objectness) + deterministic top-700 via rank counting.
// rank(i) = #{ j : v[j] > v[i]  ||  (v[j] == v[i] && j < i) }  -> matches
// jax.lax.top_k descending order with stable tie-break.
// ---------------------------------------------------------------------------
__global__ __launch_bounds__(256) void topk_kernel(
    const float* __restrict__ obj,
    float* __restrict__ scores,      // [B, 700] (tail of d_out)
    int* __restrict__ idx_out)       // [B, 700] (workspace)
{
    __shared__ float v[HW];
    const int b = blockIdx.x;
    const float* p = obj + b * HW;
    for (int i = threadIdx.x; i < HW; i += blockDim.x)
        v[i] = sigmoid_fast(p[i]);
    __syncthreads();
    for (int i = threadIdx.x; i < HW; i += blockDim.x) {
        const float vi = v[i];
        int rank = 0;
        for (int j = 0; j < HW; ++j) {
            const float vj = v[j];
            rank += (vj > vi) || ((vj == vi) && (j < i));
        }
        if (rank < TOPK) {
            scores[b * TOPK + rank]  = vi;
            idx_out[b * TOPK + rank] = i;
        }
    }
}

// ---------------------------------------------------------------------------
// Kernel B (fused): per workgroup = one batch b + one 16-row M-tile (4 final
// masks).  WMMA f32 16x16x4 (x2 for K=8) over all 256 N-tiles of the 64x64
// plane, accumulator pre-loaded with bias, sigmoid + 4-row product per lane,
// masks staged in LDS, then 4x bilinear upsample (quad-per-thread, b128
// stores) written directly to d_out.
// ---------------------------------------------------------------------------
__global__ __launch_bounds__(256) void fused_mask_kernel(
    const float* __restrict__ enc,      // [B, 8, 64, 64]
    const float* __restrict__ wts,      // [B, 36, 64, 64]
    const int* __restrict__ idx,        // [B, 700]
    float* __restrict__ mout)           // [B, 700, 256, 256]
{
    __shared__ float mlow[NG * HW];     // 4 masks x 4096 = 64 KB

    const int ktile = blockIdx.x;       // 0..174
    const int b     = blockIdx.y;       // 0..1
    const int tid   = threadIdx.x;
    const int lane  = tid & 31;
    const int wave  = tid >> 5;         // 0..7 (wave32)

    const float* encB = enc + (size_t)b * EDIM * HW;
    const float* wB   = wts + (size_t)b * (EDIM + 1) * NG * HW;
    const int*   idxB = idx + b * TOPK;

    // ---- A operand: row = lane%16, K split across lane halves -------------
    const int r    = lane & 15;
    const int half = lane >> 4;         // 0: lanes 0-15, 1: lanes 16-31
    const int kk   = half * 2;          // K base within each 4-wide slab
    const int kp   = ktile * 16 + r;    // global M row (k' index)
    const int sel  = idxB[kp >> 2];     // spatial index for this mask
    const int g    = kp & 3;            // group
    const float* wg = wB + (size_t)(g * (EDIM + 1)) * HW + sel;
    v2f A0 = { wg[(size_t)(kk + 0) * HW], wg[(size_t)(kk + 1) * HW] };  // K = kk,kk+1
    v2f A1 = { wg[(size_t)(kk + 4) * HW], wg[(size_t)(kk + 5) * HW] };  // K = kk+4,kk+5

    // ---- C init = bias, following 16x16 f32 C layout (row = v + 8*half) ---
    v8f Cinit;
#pragma unroll
    for (int v = 0; v < 8; ++v) {
        const int row = v + half * 8;
        const int kpr = ktile * 16 + row;
        const int gr  = kpr & 3;
        Cinit[v] = wB[(size_t)(gr * (EDIM + 1) + EDIM) * HW + idxB[kpr >> 2]];
    }

    const int col = lane & 15;
    const int kl0 = half * 2;           // mask index for product of v0..v3
    const int kl1 = half * 2 + 1;       // mask index for product of v4..v7

    // ---- GEMM over N: 256 tiles of 16 cols, 32 per wave -------------------
    for (int t = 0; t < 32; ++t) {
        const int n0 = (wave + 8 * t) * 16;
        const float* ep = encB + n0 + col;
        if (t + 1 < 32)                  // speculative prefetch of next tile
            __builtin_prefetch(encB + (wave + 8 * (t + 1)) * 16 + col, 0, 1);
        v2f B0 = { ep[(size_t)(kk + 0) * HW], ep[(size_t)(kk + 1) * HW] };
        v2f B1 = { ep[(size_t)(kk + 4) * HW], ep[(size_t)(kk + 5) * HW] };
        v8f c = Cinit;
        c = __builtin_amdgcn_wmma_f32_16x16x4_f32(false, A0, false, B0,
                                                  (short)0, c, false, false);
        c = __builtin_amdgcn_wmma_f32_16x16x4_f32(false, A1, false, B1,
                                                  (short)0, c, false, false);
        float s0 = sigmoid_fast(c[0]), s1 = sigmoid_fast(c[1]);
        float s2 = sigmoid_fast(c[2]), s3 = sigmoid_fast(c[3]);
        float s4 = sigmoid_fast(c[4]), s5 = sigmoid_fast(c[5]);
        float s6 = sigmoid_fast(c[6]), s7 = sigmoid_fast(c[7]);
        mlow[kl0 * HW + n0 + col] = s0 * s1 * s2 * s3;  // rows 8*half..8*half+3
        mlow[kl1 * HW + n0 + col] = s4 * s5 * s6 * s7;
    }
    __syncthreads();

    // ---- 4x bilinear upsample 64x64 -> 256x256, half-pixel centers --------
    // One thread produces a horizontal quad [4j..4j+3] from 3 x-samples
    // (constant x-weights: left pair wx={0.625,0.875} over m[j-1]..m[j],
    // right pair wx={0.125,0.375} over m[j]..m[j+1]) and emits one b128
    // store -> 512B per wave per store instruction, fully coalesced.
    float* outB = mout + ((size_t)(b * TOPK + ktile * 4)) * OUT_HW;
    for (int i = tid; i < NG * OUT_HW / 4; i += 256) {
        const int klc = i >> 14;          // mask 0..3
        const int rem = i & 16383;
        const int oy  = rem >> 6;         // output row 0..255
        const int j   = rem & 63;         // quad index 0..63 (ox = 4j)

        const float fy  = (oy + 0.5f) * 0.25f - 0.5f;
        const float y0f = floorf(fy);
        const float wy  = fy - y0f;
        int y0 = (int)y0f;
        int y1 = min(max(y0 + 1, 0), 63);
        y0 = min(max(y0, 0), 63);

        const int jm1 = max(j - 1, 0);
        const int jp1 = min(j + 1, 63);

        const float* m  = mlow + klc * HW;
        const float* r0 = m + y0 * 64;
        const float* r1 = m + y1 * 64;

        // vertical lerp of the three x-samples
        const float a0 = r0[jm1], a1 = r1[jm1];
        const float b0 = r0[j],   b1 = r1[j];
        const float c0 = r0[jp1], c1 = r1[jp1];
        const float A = a0 + wy * (a1 - a0);
        const float B = b0 + wy * (b1 - b0);
        const float C = c0 + wy * (c1 - c0);

        v4f q;
        q.x = A + 0.625f * (B - A);       // ox = 4j+0
        q.y = A + 0.875f * (B - A);       // ox = 4j+1
        q.z = B + 0.125f * (C - B);       // ox = 4j+2
        q.w = B + 0.375f * (C - B);       // ox = 4j+3
        *(v4f*)(outB + (size_t)klc * OUT_HW + oy * 256 + 4 * j) = q;
    }
}

// ---------------------------------------------------------------------------
extern "C" void kernel_launch(void* const* d_in, const int* in_sizes, int n_in,
                              void* d_out, int out_size, void* d_ws, size_t ws_size,
                              hipStream_t stream) {
    const float* obj = (const float*)d_in[0];   // [2,1,64,64]
    const float* enc = (const float*)d_in[1];   // [2,8,64,64]
    const float* wts = (const float*)d_in[2];   // [2,36,64,64]

    float* out_m      = (float*)d_out;                             // [2,700,256,256]
    float* out_scores = out_m + (size_t)BDIM * TOPK * OUT_HW;      // [2,700]
    int*   idx_ws     = (int*)d_ws;                                // [2,700]

    topk_kernel<<<BDIM, 256, 0, stream>>>(obj, out_scores, idx_ws);

    dim3 grid(TOPK / NG, BDIM);   // (175, 2)
    fused_mask_kernel<<<grid, 256, 0, stream>>>(enc, wts, idx_ws, out_m);
}